// resampler_22926535426674
// MI455X (gfx1250) — compile-verified
//
#include <hip/hip_runtime.h>
#include <math.h>

// Sinkhorn OT resampler for MI455X (gfx1250), f32 end-to-end.
// Strategy: one 64MB cost matrix, L2-resident (192MB L2); Gram + final
// transport matmul via V_WMMA_F32_16X16X4_F32; Sinkhorn softmins as single
// streaming passes with block reductions; per-batch eps annealing + global
// early-stop flag replicated on device across 99 fixed launches.

#define B_ 16
#define N_ 1024
#define D_ 32
#define EPS_F 0.1f
#define SF_F 0.5625f                  // SCALING^2 = 0.75^2
#define THR_F 0.001f
#define MAXIT 100
#define LOGN_F 6.9314718055994531f    // log(1024)
#define LB_F (-6.9314718055994531f)   // uniform log-weight = -log(N)

typedef __attribute__((ext_vector_type(2))) float v2f;
typedef __attribute__((ext_vector_type(8))) float v8f;

// ---- float <-> ordered-uint keys for atomic min/max over signed floats ----
__device__ __forceinline__ unsigned fkey(float f){
  unsigned u = __float_as_uint(f);
  return (u & 0x80000000u) ? ~u : (u | 0x80000000u);
}
__device__ __forceinline__ float funkey(unsigned u){
  return (u & 0x80000000u) ? __uint_as_float(u & 0x7FFFFFFFu)
                           : __uint_as_float(~u);
}

// ---- block (256 threads = 8 wave32) reduction of 4 lanes of data ----
__device__ __forceinline__ void blk_red4(float v[4], float* lds, int is_max){
  int lane = threadIdx.x & 31, wid = threadIdx.x >> 5;
  #pragma unroll
  for (int k = 0; k < 4; ++k){
    float x = v[k];
    #pragma unroll
    for (int o = 16; o > 0; o >>= 1){
      float y = __shfl_xor(x, o, 32);
      x = is_max ? fmaxf(x, y) : (x + y);
    }
    if (lane == 0) lds[k*8 + wid] = x;
  }
  __syncthreads();
  #pragma unroll
  for (int k = 0; k < 4; ++k){
    float x = lds[k*8];
    #pragma unroll
    for (int w = 1; w < 8; ++w){
      float y = lds[k*8 + w];
      x = is_max ? fmaxf(x, y) : (x + y);
    }
    v[k] = x;
  }
  __syncthreads();
}

// ---- per-batch mean / std / diameter -> normalization scale ----
__global__ void k_stats(const float* __restrict__ P, float* __restrict__ mean,
                        float* __restrict__ inv_scale,
                        unsigned* __restrict__ umax, unsigned* __restrict__ umin){
  int b = blockIdx.x, t = threadIdx.x;
  int d = t & 31, c = t >> 5;
  float s = 0.f, ss = 0.f;
  for (int n = c; n < N_; n += 8){
    float v = P[((size_t)b*N_ + n)*D_ + d];
    s += v; ss += v*v;
  }
  __shared__ float ls[256], lss[256];
  ls[t] = s; lss[t] = ss;
  __syncthreads();
  if (t < 32){
    float S = 0.f, SS = 0.f;
    #pragma unroll
    for (int cc = 0; cc < 8; ++cc){ S += ls[cc*32 + t]; SS += lss[cc*32 + t]; }
    float m  = S / (float)N_;
    float var = SS / (float)N_ - m*m;
    float sd  = sqrtf(fmaxf(var, 0.f));
    mean[b*D_ + t] = m;
    float mx = sd;
    #pragma unroll
    for (int o = 16; o > 0; o >>= 1) mx = fmaxf(mx, __shfl_xor(mx, o, 32));
    if (t == 0){
      float diam = (mx == 0.f) ? 1.f : mx;
      inv_scale[b] = 1.f / (diam * sqrtf((float)D_));
      umax[b] = 0u;
      umin[b] = 0xFFFFFFFFu;
    }
  }
}

// ---- normalize particles, row |x|^2, log-weights, global min/max for eps0 ----
__global__ void k_normalize(const float* __restrict__ P, const float* __restrict__ W,
                            const float* __restrict__ mean, const float* __restrict__ inv_scale,
                            float* __restrict__ sx, float* __restrict__ x2,
                            float* __restrict__ logw,
                            unsigned* __restrict__ umax, unsigned* __restrict__ umin){
  int b = blockIdx.y;
  int n = blockIdx.x * 8 + (threadIdx.x >> 5);
  int d = threadIdx.x & 31;
  float v = (P[((size_t)b*N_ + n)*D_ + d] - mean[b*D_ + d]) * inv_scale[b];
  sx[((size_t)b*N_ + n)*D_ + d] = v;
  float s2 = v*v, mx = v, mn = v;
  #pragma unroll
  for (int o = 16; o > 0; o >>= 1){
    s2 += __shfl_xor(s2, o, 32);
    mx  = fmaxf(mx, __shfl_xor(mx, o, 32));
    mn  = fminf(mn, __shfl_xor(mn, o, 32));
  }
  if (d == 0){
    x2[b*N_ + n]   = s2;
    logw[b*N_ + n] = logf(W[b*N_ + n]);
    atomicMax(&umax[b], fkey(mx));
    atomicMin(&umin[b], fkey(mn));
  }
}

// ---- eps0 = (max-min)^2 per batch; reset iteration state ----
__global__ void k_prep(const unsigned* __restrict__ umax, const unsigned* __restrict__ umin,
                       float* __restrict__ re, unsigned* __restrict__ md_ay,
                       unsigned* __restrict__ md_bx, unsigned* __restrict__ flag){
  int t = threadIdx.x;
  if (t < B_){
    float dm = funkey(umax[t]) - funkey(umin[t]);
    re[t] = dm * dm;
    md_ay[t] = 0u; md_bx[t] = 0u;
  }
  if (t == 0) *flag = 1u;
}

// ---- C[b,i,j] = max(0.5*(|xi|^2+|xj|^2) - <xi,xj>, 0) via f32 WMMA Gram ----
__global__ void k_cost(const float* __restrict__ sx, const float* __restrict__ x2,
                       float* __restrict__ C){
  int b  = blockIdx.y;
  int ti = blockIdx.x >> 6, tj = blockIdx.x & 63;
  int L  = threadIdx.x;
  int m  = L & 15, hi = L >> 4;
  int i0 = ti * 16, j0 = tj * 16;
  const float* Ai = sx + ((size_t)b*N_ + i0 + m)*D_ + 2*hi;  // A: M=m, K=v+2*hi
  const float* Bj = sx + ((size_t)b*N_ + j0 + m)*D_ + 2*hi;  // B: N=m, K=v+2*hi
  v8f acc = {0.f,0.f,0.f,0.f,0.f,0.f,0.f,0.f};
  #pragma unroll
  for (int k0 = 0; k0 < D_; k0 += 4){
    v2f a, bb;
    a.x  = Ai[k0];  a.y  = Ai[k0 + 1];
    bb.x = Bj[k0];  bb.y = Bj[k0 + 1];
    acc = __builtin_amdgcn_wmma_f32_16x16x4_f32(false, a, false, bb,
                                                (short)0, acc, false, false);
  }
  float x2j = x2[b*N_ + j0 + m];
  #pragma unroll
  for (int r = 0; r < 8; ++r){
    int i = i0 + r + 8*hi;                         // C/D: row r+8*hi, col m
    float c = 0.5f*(x2[b*N_ + i] + x2j) - acc[r];
    C[((size_t)b*N_ + i)*N_ + j0 + m] = fmaxf(c, 0.f);
  }
}

// ---- initial potentials: softmin(eps0, C, logw) and softmin(eps0, C, -logN) ----
__global__ void k_pot_init(const float* __restrict__ C, const float* __restrict__ logw,
                           const float* __restrict__ re, float* __restrict__ buf0){
  __shared__ float lds[32];
  const int BN = B_ * N_;
  int b = blockIdx.x >> 10, i = blockIdx.x & (N_ - 1);
  float eps = re[b], inv = 1.f / eps;
  const float* Crow = C + ((size_t)b*N_ + i) * N_;
  int j0 = threadIdx.x * 4;
  float4 c4  = *(const float4*)(Crow + j0);
  float4 lw4 = *(const float4*)(logw + b*N_ + j0);
  float cc[4] = {c4.x, c4.y, c4.z, c4.w};
  float lw[4] = {lw4.x, lw4.y, lw4.z, lw4.w};
  float t0[4], t1[4];
  float m[4] = {-__builtin_inff(), -__builtin_inff(), -__builtin_inff(), -__builtin_inff()};
  #pragma unroll
  for (int e = 0; e < 4; ++e){
    float ci = cc[e] * inv;
    t0[e] = lw[e] - ci;
    t1[e] = LB_F  - ci;
    m[0] = fmaxf(m[0], t0[e]);
    m[1] = fmaxf(m[1], t1[e]);
  }
  blk_red4(m, lds, 1);
  float s[4] = {0.f, 0.f, 0.f, 0.f};
  #pragma unroll
  for (int e = 0; e < 4; ++e){
    s[0] += expf(t0[e] - m[0]);
    s[1] += expf(t1[e] - m[1]);
  }
  blk_red4(s, lds, 0);
  if (threadIdx.x == 0){
    float sa = -eps * (logf(s[0]) + m[0]);
    float sb = -eps * (logf(s[1]) + m[1]);
    buf0[0*BN + b*N_ + i] = sa;   // a_y0
    buf0[1*BN + b*N_ + i] = sb;   // b_x0
    buf0[2*BN + b*N_ + i] = sa;   // a_x0
    buf0[3*BN + b*N_ + i] = sb;   // b_y0
  }
}

// ---- one Sinkhorn iteration: 4 fused softmins over one C row scan ----
__global__ void k_iter(const float* __restrict__ C, const float* __restrict__ logw,
                       const float* __restrict__ re,
                       const float* __restrict__ in, float* __restrict__ out,
                       unsigned* __restrict__ md_ay, unsigned* __restrict__ md_bx,
                       const unsigned* __restrict__ flag){
  __shared__ float lds[32];
  const int BN = B_ * N_;
  int b = blockIdx.x >> 10, i = blockIdx.x & (N_ - 1);
  if (*flag == 0u){                       // converged: keep ping-pong consistent
    if (threadIdx.x == 0){
      #pragma unroll
      for (int q = 0; q < 4; ++q) out[q*BN + b*N_ + i] = in[q*BN + b*N_ + i];
    }
    return;
  }
  float eps = re[b], inv = 1.f / eps;
  const float* Crow = C + ((size_t)b*N_ + i) * N_;
  const float* ay = in + 0*BN + b*N_;
  const float* bx = in + 1*BN + b*N_;
  const float* ax = in + 2*BN + b*N_;
  const float* by = in + 3*BN + b*N_;
  int j0 = threadIdx.x * 4;
  float4 c4  = *(const float4*)(Crow + j0);
  float4 lw4 = *(const float4*)(logw + b*N_ + j0);
  float4 bx4 = *(const float4*)(bx + j0);
  float4 ay4 = *(const float4*)(ay + j0);
  float4 ax4 = *(const float4*)(ax + j0);
  float4 by4 = *(const float4*)(by + j0);
  float cc[4] = {c4.x,c4.y,c4.z,c4.w};
  float lw[4] = {lw4.x,lw4.y,lw4.z,lw4.w};
  float vbx[4] = {bx4.x,bx4.y,bx4.z,bx4.w};
  float vay[4] = {ay4.x,ay4.y,ay4.z,ay4.w};
  float vax[4] = {ax4.x,ax4.y,ax4.z,ax4.w};
  float vby[4] = {by4.x,by4.y,by4.z,by4.w};
  float t1[4], t2[4], t3[4], t4[4];
  float m[4] = {-__builtin_inff(), -__builtin_inff(), -__builtin_inff(), -__builtin_inff()};
  #pragma unroll
  for (int e = 0; e < 4; ++e){
    float ci = cc[e] * inv;
    t1[e] = lw[e] + vbx[e]*inv - ci;   // at_y: la + b_x/re
    t2[e] = LB_F  + vay[e]*inv - ci;   // bt_x: lb + a_y/re
    t3[e] = lw[e] + vax[e]*inv - ci;   // at_x: la + a_x/re
    t4[e] = LB_F  + vby[e]*inv - ci;   // bt_y: lb + b_y/re
    m[0] = fmaxf(m[0], t1[e]); m[1] = fmaxf(m[1], t2[e]);
    m[2] = fmaxf(m[2], t3[e]); m[3] = fmaxf(m[3], t4[e]);
  }
  blk_red4(m, lds, 1);
  float s[4] = {0.f, 0.f, 0.f, 0.f};
  #pragma unroll
  for (int e = 0; e < 4; ++e){
    s[0] += expf(t1[e] - m[0]); s[1] += expf(t2[e] - m[1]);
    s[2] += expf(t3[e] - m[2]); s[3] += expf(t4[e] - m[3]);
  }
  blk_red4(s, lds, 0);
  if (threadIdx.x == 0){
    float s1 = -eps*(logf(s[0]) + m[0]);
    float s2 = -eps*(logf(s[1]) + m[1]);
    float s3 = -eps*(logf(s[2]) + m[2]);
    float s4 = -eps*(logf(s[3]) + m[3]);
    float o1 = ay[i], o2 = bx[i], o3 = ax[i], o4 = by[i];
    float n1 = 0.5f*(o1 + s1), n2 = 0.5f*(o2 + s2);
    float n3 = 0.5f*(o3 + s3), n4 = 0.5f*(o4 + s4);
    out[0*BN + b*N_ + i] = n1;
    out[1*BN + b*N_ + i] = n2;
    out[2*BN + b*N_ + i] = n3;
    out[3*BN + b*N_ + i] = n4;
    atomicMax(md_ay + b, __float_as_uint(fabsf(n1 - o1)));  // nonneg: raw bits ordered
    atomicMax(md_bx + b, __float_as_uint(fabsf(n2 - o2)));
  }
}

// ---- eps anneal + global continue flag (all(cont), like jax while_loop) ----
__global__ void k_update(float* __restrict__ re, unsigned* __restrict__ md_ay,
                         unsigned* __restrict__ md_bx, unsigned* __restrict__ flag){
  int t = threadIdx.x;
  if (*flag == 0u) return;
  int g = 1;
  if (t < B_){
    float d1 = __uint_as_float(md_ay[t]);
    float d2 = __uint_as_float(md_bx[t]);
    int local = (d1 > THR_F) || (d2 > THR_F);
    float ne = fmaxf(re[t] * SF_F, EPS_F);
    g = (ne < re[t]) || local;
    re[t] = ne;
    md_ay[t] = 0u; md_bx[t] = 0u;
  }
  int allg = __all(g);
  if (t == 0) *flag = allg ? 1u : 0u;
}

// ---- final potentials at eps=EPS: alpha = f_a_y, beta = f_b_x ----
__global__ void k_final(const float* __restrict__ C, const float* __restrict__ logw,
                        const float* __restrict__ buf,
                        float* __restrict__ alpha, float* __restrict__ beta){
  __shared__ float lds[32];
  const int BN = B_ * N_;
  int b = blockIdx.x >> 10, i = blockIdx.x & (N_ - 1);
  const float inv = 1.f / EPS_F;
  const float* Crow = C + ((size_t)b*N_ + i) * N_;
  const float* ay = buf + 0*BN + b*N_;
  const float* bx = buf + 1*BN + b*N_;
  int j0 = threadIdx.x * 4;
  float4 c4  = *(const float4*)(Crow + j0);
  float4 lw4 = *(const float4*)(logw + b*N_ + j0);
  float4 bx4 = *(const float4*)(bx + j0);
  float4 ay4 = *(const float4*)(ay + j0);
  float cc[4] = {c4.x,c4.y,c4.z,c4.w};
  float lw[4] = {lw4.x,lw4.y,lw4.z,lw4.w};
  float vb[4] = {bx4.x,bx4.y,bx4.z,bx4.w};
  float va[4] = {ay4.x,ay4.y,ay4.z,ay4.w};
  float t0[4], t1[4];
  float m[4] = {-__builtin_inff(), -__builtin_inff(), -__builtin_inff(), -__builtin_inff()};
  #pragma unroll
  for (int e = 0; e < 4; ++e){
    float ci = cc[e] * inv;
    t0[e] = lw[e] + vb[e]*inv - ci;   // alpha: la + b_x/eps
    t1[e] = LB_F  + va[e]*inv - ci;   // beta : lb + a_y/eps
    m[0] = fmaxf(m[0], t0[e]);
    m[1] = fmaxf(m[1], t1[e]);
  }
  blk_red4(m, lds, 1);
  float s[4] = {0.f, 0.f, 0.f, 0.f};
  #pragma unroll
  for (int e = 0; e < 4; ++e){
    s[0] += expf(t0[e] - m[0]);
    s[1] += expf(t1[e] - m[1]);
  }
  blk_red4(s, lds, 0);
  if (threadIdx.x == 0){
    alpha[b*N_ + i] = -EPS_F * (logf(s[0]) + m[0]);
    beta [b*N_ + i] = -EPS_F * (logf(s[1]) + m[1]);
  }
}

// ---- column logsumexp of transport logits (C symmetric -> scan row j) ----
// w_j = log n + logw_j - logsumexp_i((alpha_i - C_ij)/eps)   (beta cancels)
__global__ void k_colsum(const float* __restrict__ C, const float* __restrict__ alpha,
                         const float* __restrict__ logw, float* __restrict__ wcol){
  __shared__ float lds[32];
  int b = blockIdx.x >> 10, j = blockIdx.x & (N_ - 1);
  const float inv = 1.f / EPS_F;
  const float* Crow = C + ((size_t)b*N_ + j) * N_;
  int i0 = threadIdx.x * 4;
  float4 c4 = *(const float4*)(Crow + i0);
  float4 a4 = *(const float4*)(alpha + b*N_ + i0);
  float cc[4] = {c4.x,c4.y,c4.z,c4.w};
  float aa[4] = {a4.x,a4.y,a4.z,a4.w};
  float t[4];
  float m[4] = {-__builtin_inff(), -__builtin_inff(), -__builtin_inff(), -__builtin_inff()};
  #pragma unroll
  for (int e = 0; e < 4; ++e){
    t[e] = (aa[e] - cc[e]) * inv;
    m[0] = fmaxf(m[0], t[e]);
  }
  blk_red4(m, lds, 1);
  float s[4] = {0.f, 0.f, 0.f, 0.f};
  #pragma unroll
  for (int e = 0; e < 4; ++e) s[0] += expf(t[e] - m[0]);
  blk_red4(s, lds, 0);
  if (threadIdx.x == 0){
    float lse = logf(s[0]) + m[0];
    wcol[b*N_ + j] = LOGN_F + logw[b*N_ + j] - lse;
  }
}

// ---- out = T @ particles with T built on the fly, via f32 WMMA ----
__global__ void k_resample(const float* __restrict__ C, const float* __restrict__ P,
                           const float* __restrict__ alpha, const float* __restrict__ wcol,
                           float* __restrict__ out){
  int b  = blockIdx.y;
  int i0 = blockIdx.x * 16;
  int L  = threadIdx.x;
  int m  = L & 15, hi = L >> 4;
  const float invE = 1.f / EPS_F;
  float aterm = alpha[b*N_ + i0 + m] * invE;
  const float* Crow = C + ((size_t)b*N_ + i0 + m) * N_;
  const float* wc = wcol + b*N_;
  const float* Pb = P + (size_t)b*N_*D_;
  v8f acc0 = {0.f,0.f,0.f,0.f,0.f,0.f,0.f,0.f};
  v8f acc1 = {0.f,0.f,0.f,0.f,0.f,0.f,0.f,0.f};
  for (int j0 = 0; j0 < N_; j0 += 4){
    int jA = j0 + 2*hi;                              // K = v + 2*hi
    v2f a, b0, b1;
    a.x = expf(aterm + wc[jA]     - Crow[jA]    *invE);
    a.y = expf(aterm + wc[jA + 1] - Crow[jA + 1]*invE);
    b0.x = Pb[(size_t)jA      *D_ + m];
    b0.y = Pb[(size_t)(jA + 1)*D_ + m];
    b1.x = Pb[(size_t)jA      *D_ + 16 + m];
    b1.y = Pb[(size_t)(jA + 1)*D_ + 16 + m];
    acc0 = __builtin_amdgcn_wmma_f32_16x16x4_f32(false, a, false, b0,
                                                 (short)0, acc0, false, false);
    acc1 = __builtin_amdgcn_wmma_f32_16x16x4_f32(false, a, false, b1,
                                                 (short)0, acc1, false, false);
  }
  #pragma unroll
  for (int r = 0; r < 8; ++r){
    int i = i0 + r + 8*hi;
    out[((size_t)b*N_ + i)*D_ + m]      = acc0[r];
    out[((size_t)b*N_ + i)*D_ + 16 + m] = acc1[r];
  }
}

// ---- probs = 1/N, index ramp ----
__global__ void k_outputs(float* __restrict__ out_probs, float* __restrict__ out_idx){
  int g = blockIdx.x * 256 + threadIdx.x;
  out_probs[g] = 1.0f / (float)N_;
  out_idx[g]   = (float)g;   // arange(N) + N*arange(B) flattened == g (exact in f32)
}

extern "C" void kernel_launch(void* const* d_in, const int* in_sizes, int n_in,
                              void* d_out, int out_size, void* d_ws, size_t ws_size,
                              hipStream_t stream){
  (void)in_sizes; (void)n_in; (void)out_size; (void)ws_size;
  const float* P = (const float*)d_in[0];
  const float* W = (const float*)d_in[1];
  float* out = (float*)d_out;
  float* ws  = (float*)d_ws;

  const size_t BN = (size_t)B_ * N_;
  float* Cm   = ws;                                // B*N*N  (64 MB, L2-resident)
  float* sx   = Cm   + (size_t)B_*N_*N_;           // B*N*D
  float* x2   = sx   + BN*D_;                      // B*N
  float* logw = x2   + BN;                         // B*N
  float* buf0 = logw + BN;                         // 4*B*N
  float* buf1 = buf0 + 4*BN;                       // 4*B*N
  float* alpha= buf1 + 4*BN;                       // B*N
  float* beta = alpha+ BN;                         // B*N
  float* wcol = beta + BN;                         // B*N
  float* mean = wcol + BN;                         // B*D
  float* invs = mean + (size_t)B_*D_;              // B
  float* re   = invs + B_;                         // B
  unsigned* umax = (unsigned*)(re + B_);           // B
  unsigned* umin = umax + B_;                      // B
  unsigned* mday = umin + B_;                      // B
  unsigned* mdbx = mday + B_;                      // B
  unsigned* flag = mdbx + B_;                      // 1

  k_stats    <<<B_, 256, 0, stream>>>(P, mean, invs, umax, umin);
  k_normalize<<<dim3(N_/8, B_), 256, 0, stream>>>(P, W, mean, invs, sx, x2, logw, umax, umin);
  k_prep     <<<1, 32, 0, stream>>>(umax, umin, re, mday, mdbx, flag);
  k_cost     <<<dim3((N_/16)*(N_/16), B_), 32, 0, stream>>>(sx, x2, Cm);
  k_pot_init <<<B_*N_, 256, 0, stream>>>(Cm, logw, re, buf0);

  for (int t = 0; t < MAXIT - 1; ++t){             // 99 body iterations max
    float* pin  = (t & 1) ? buf1 : buf0;
    float* pout = (t & 1) ? buf0 : buf1;
    k_iter   <<<B_*N_, 256, 0, stream>>>(Cm, logw, re, pin, pout, mday, mdbx, flag);
    k_update <<<1, 32, 0, stream>>>(re, mday, mdbx, flag);
  }
  // after 99 iterations (odd), final potentials live in buf1
  k_final   <<<B_*N_, 256, 0, stream>>>(Cm, logw, buf1, alpha, beta);
  k_colsum  <<<B_*N_, 256, 0, stream>>>(Cm, alpha, logw, wcol);
  k_resample<<<dim3(N_/16, B_), 32, 0, stream>>>(Cm, P, alpha, wcol, out);
  k_outputs <<<(int)(BN/256), 256, 0, stream>>>(out + BN*D_, out + BN*D_ + BN);
}